// GNN_44418551775999
// MI455X (gfx1250) — compile-verified
//
#include <hip/hip_runtime.h>
#include <hip/hip_bf16.h>

#define TOT   65536
#define NB    128
#define NN    512
#define FIN   128
#define EE    1048576

typedef __attribute__((ext_vector_type(2))) float v2f;
typedef __attribute__((ext_vector_type(8))) float v8f;

__device__ inline v8f wmma4(v2f a, v2f b, v8f c) {
    // V_WMMA_F32_16X16X4_F32 : D = A(16x4) x B(4x16) + C, full f32
    return __builtin_amdgcn_wmma_f32_16x16x4_f32(
        false, a, false, b, (short)0, c, false, false);
}

__device__ inline float fast_sigmoid(float x) {
    // v_rcp_f32 (~1 ulp) instead of IEEE divide expansion
    return __builtin_amdgcn_rcpf(1.0f + __expf(-x));
}

// ---------------------------------------------------------------------------
// Row-major GEMM: out[M x NOUT] = act(X)[M x K] @ W[K x NOUT] (+bias)
// One wave per 16-row strip, NOUT/16 accumulators reuse the A fragment.
// RELU applies relu to X on load (folds the previous layer's activation).
// ---------------------------------------------------------------------------
template <int K, int NOUT, bool RELU>
__global__ void gemm_wmma(const float* __restrict__ X, const float* __restrict__ W,
                          const float* __restrict__ bias, float* __restrict__ out, int M)
{
    const int lane = threadIdx.x & 31;
    const int wave = blockIdx.x * (blockDim.x >> 5) + (threadIdx.x >> 5);
    const int m0 = wave * 16;
    if (m0 >= M) return;
    const int r  = lane & 15;          // row-in-tile (A) / col-in-tile (B)
    const int kq = (lane >> 4) * 2;    // K sub-offset per lane half
    constexpr int NT = NOUT / 16;
    const v8f vzero = {};
    v8f acc[NT];
#pragma unroll
    for (int n = 0; n < NT; ++n) acc[n] = vzero;

    const float* xrow = X + (size_t)(m0 + r) * K;
    for (int k = 0; k < K; k += 4) {
        v2f a = *(const v2f*)(xrow + k + kq);      // 8B-aligned b64 load
        if (RELU) { a.x = fmaxf(a.x, 0.0f); a.y = fmaxf(a.y, 0.0f); }
#pragma unroll
        for (int n = 0; n < NT; ++n) {
            const int col = n * 16 + r;
            v2f bf;
            bf.x = W[(size_t)(k + kq) * NOUT + col];
            bf.y = W[(size_t)(k + kq + 1) * NOUT + col];
            acc[n] = wmma4(a, bf, acc[n]);
        }
    }
    const int mbase = m0 + (lane >> 4) * 8;   // C/D layout: vgpr rr -> M=rr / M=rr+8
#pragma unroll
    for (int n = 0; n < NT; ++n) {
        const int col = n * 16 + r;
        const float bb = bias ? bias[col] : 0.0f;
#pragma unroll
        for (int rr = 0; rr < 8; ++rr)
            out[(size_t)(mbase + rr) * NOUT + col] = acc[n][rr] + bb;
    }
}

// ---------------------------------------------------------------------------
// GCN support kernels
// ---------------------------------------------------------------------------
__global__ void fill_f32(float* p, float v, int n) {
    int i = blockIdx.x * blockDim.x + threadIdx.x;
    if (i < n) p[i] = v;
}
__global__ void fill_u32(unsigned int* p, unsigned int v, int n) {
    int i = blockIdx.x * blockDim.x + threadIdx.x;
    if (i < n) p[i] = v;
}
__global__ void deg_count(const int* __restrict__ dst, float* __restrict__ deg, int ne) {
    int e = blockIdx.x * blockDim.x + threadIdx.x;
    if (e < ne) atomicAdd(&deg[dst[e]], 1.0f);
}
// dis = rsqrt(deg), inv = 1/deg (exact; cold pass)
__global__ void deg_finalize(const float* __restrict__ deg, float* __restrict__ dis,
                             float* __restrict__ inv, int n) {
    int i = blockIdx.x * blockDim.x + threadIdx.x;
    if (i < n) { float d = deg[i]; dis[i] = rsqrtf(d); inv[i] = 1.0f / d; }
}
// out = g * inv_deg + bias  (self-loop term; float4 streaming, overwrites out)
template <int F>
__global__ void gcn_self(const float* __restrict__ g, const float* __restrict__ inv,
                         const float* __restrict__ bias, float* __restrict__ out, int M)
{
    int i = blockIdx.x * blockDim.x + threadIdx.x;       // float4 index
    if (i >= M * (F / 4)) return;
    int node = i / (F / 4);
    int c = (i - node * (F / 4)) * 4;
    float iv = inv[node];
    float4 gv = *(const float4*)(g + (size_t)i * 4);
    float4 o;
    o.x = gv.x * iv + bias[c + 0];
    o.y = gv.y * iv + bias[c + 1];
    o.z = gv.z * iv + bias[c + 2];
    o.w = gv.w * iv + bias[c + 3];
    *(float4*)(out + (size_t)i * 4) = o;
}
// out[dst] += g[src] * dis[src] * dis[dst]  (float4 per thread)
template <int F>
__global__ void gcn_agg(const int* __restrict__ src, const int* __restrict__ dst,
                        const float* __restrict__ g, const float* __restrict__ dis,
                        float* __restrict__ out, int ne)
{
    constexpr int C = F / 4;
    int id = blockIdx.x * blockDim.x + threadIdx.x;
    if (id >= ne * C) return;
    int e = id / C;
    int c = (id - e * C) * 4;
    int s = src[e], d = dst[e];
    float coef = dis[s] * dis[d];
    float4 gv = *(const float4*)(g + (size_t)s * F + c);
    float* o = out + (size_t)d * F + c;
    atomicAdd(o + 0, gv.x * coef);
    atomicAdd(o + 1, gv.y * coef);
    atomicAdd(o + 2, gv.z * coef);
    atomicAdd(o + 3, gv.w * coef);
}

// ---------------------------------------------------------------------------
// Similarity: S[b] = d1[b] @ d2[b]^T (512x512, K=32). Each wave computes a
// 16x32 strip (two accumulators share one A fragment). 512 waves per graph.
// ---------------------------------------------------------------------------
__device__ inline void sim_strip(const float* __restrict__ af1, const float* __restrict__ af2,
                                 int b, int tm, int n0, int lane, v8f& a0, v8f& a1)
{
    const int r  = lane & 15;
    const int kq = (lane >> 4) * 2;
    const float* arow  = af1 + ((size_t)b * NN + tm * 16 + r) * 32;
    const float* brow0 = af2 + ((size_t)b * NN + n0 + r) * 32;       // B[k][n]=d2[n][k]
    const float* brow1 = af2 + ((size_t)b * NN + n0 + 16 + r) * 32;
    a0 = (v8f){}; a1 = (v8f){};
#pragma unroll
    for (int k = 0; k < 32; k += 4) {
        v2f a  = *(const v2f*)(arow  + k + kq);
        v2f b0 = *(const v2f*)(brow0 + k + kq);
        v2f b1 = *(const v2f*)(brow1 + k + kq);
        a0 = wmma4(a, b0, a0);
        a1 = wmma4(a, b1, a1);
    }
}

__global__ void sim_minmax(const float* __restrict__ af1, const float* __restrict__ af2,
                           unsigned int* __restrict__ gmin, unsigned int* __restrict__ gmax)
{
    const int b = blockIdx.y;
    const int lane = threadIdx.x & 31;
    const int tileId = blockIdx.x * (blockDim.x >> 5) + (threadIdx.x >> 5); // 0..511
    const int tm = tileId >> 4, n0 = (tileId & 15) * 32;
    v8f a0, a1;
    sim_strip(af1, af2, b, tm, n0, lane, a0, a1);
    float mn = 3.4e38f, mx = -3.4e38f;
#pragma unroll
    for (int rr = 0; rr < 8; ++rr) {
        float s0 = fast_sigmoid(a0[rr]);
        float s1 = fast_sigmoid(a1[rr]);
        mn = fminf(mn, fminf(s0, s1));
        mx = fmaxf(mx, fmaxf(s0, s1));
    }
    for (int off = 16; off; off >>= 1) {
        mn = fminf(mn, __shfl_xor(mn, off, 32));
        mx = fmaxf(mx, __shfl_xor(mx, off, 32));
    }
    if (lane == 0) {  // sigmoid > 0: uint ordering == float ordering
        atomicMin(&gmin[b], __float_as_uint(mn));
        atomicMax(&gmax[b], __float_as_uint(mx));
    }
}

__global__ void sim_hist(const float* __restrict__ af1, const float* __restrict__ af2,
                         const unsigned int* __restrict__ gmin, const unsigned int* __restrict__ gmax,
                         unsigned int* __restrict__ ghist)
{
    __shared__ unsigned int lh[16];
    const int b = blockIdx.y;
    if (threadIdx.x < 16) lh[threadIdx.x] = 0;
    __syncthreads();
    const int lane = threadIdx.x & 31;
    const int tileId = blockIdx.x * (blockDim.x >> 5) + (threadIdx.x >> 5);
    const int tm = tileId >> 4, n0 = (tileId & 15) * 32;
    v8f a0, a1;
    sim_strip(af1, af2, b, tm, n0, lane, a0, a1);
    const float mn  = __uint_as_float(gmin[b]);
    const float mx  = __uint_as_float(gmax[b]);
    const float rng = (mx > mn) ? (mx - mn) : 1.0f;
    const float sc  = 16.0f / rng;
#pragma unroll
    for (int rr = 0; rr < 8; ++rr) {
        int b0 = (int)floorf((fast_sigmoid(a0[rr]) - mn) * sc);
        int b1 = (int)floorf((fast_sigmoid(a1[rr]) - mn) * sc);
        b0 = b0 < 0 ? 0 : (b0 > 15 ? 15 : b0);
        b1 = b1 < 0 ? 0 : (b1 > 15 ? 15 : b1);
        atomicAdd(&lh[b0], 1u);
        atomicAdd(&lh[b1], 1u);
    }
    __syncthreads();
    if (threadIdx.x < 16) atomicAdd(&ghist[b * 16 + threadIdx.x], lh[threadIdx.x]);
}

// ---------------------------------------------------------------------------
// Attention pooling: one block (256 thr) per graph; lane == feature (F3=32)
// ---------------------------------------------------------------------------
__global__ void att_pool(const float* __restrict__ h, const float* __restrict__ hAtt,
                         float* __restrict__ pooled)
{
    const int b = blockIdx.x;
    __shared__ float red[8][32];
    __shared__ float t[32];
    const int f = threadIdx.x & 31;
    const int w = threadIdx.x >> 5;   // 0..7
    float acc = 0.f;
    for (int n = w; n < NN; n += 8)
        acc += hAtt[((size_t)b * NN + n) * 32 + f];
    red[w][f] = acc;
    __syncthreads();
    if (w == 0) {
        float g = 0.f;
        for (int i = 0; i < 8; ++i) g += red[i][f];
        t[f] = tanhf(g / (float)NN);
    }
    __syncthreads();
    const float tf = t[f];
    float pool = 0.f;
    for (int n = w; n < NN; n += 8) {
        float hv = h[((size_t)b * NN + n) * 32 + f];
        float prod = hv * tf;
        for (int off = 16; off; off >>= 1) prod += __shfl_xor(prod, off, 32);
        float s = 1.0f / (1.0f + __expf(-prod));
        pool += s * hv;
    }
    red[w][f] = pool;
    __syncthreads();
    if (w == 0) {
        float p = 0.f;
        for (int i = 0; i < 8; ++i) p += red[i][f];
        pooled[b * 32 + f] = p;
    }
}

// ---------------------------------------------------------------------------
// NTN + histogram normalize + FC head: one block (64 thr) per graph pair
// ---------------------------------------------------------------------------
__global__ void ntn_final(const float* __restrict__ p1g, const float* __restrict__ p2g,
                          const unsigned int* __restrict__ ghist,
                          const float* __restrict__ ntnW, const float* __restrict__ ntnV,
                          const float* __restrict__ ntnb,
                          const float* __restrict__ fc1W, const float* __restrict__ fc1b,
                          const float* __restrict__ scW, const float* __restrict__ scb,
                          float* __restrict__ out)
{
    const int b = blockIdx.x;
    __shared__ float p1[32], p2[32], feat[32], h1[16];
    const int t = threadIdx.x;
    if (t < 32) { p1[t] = p1g[b * 32 + t]; p2[t] = p2g[b * 32 + t]; }
    __syncthreads();
    if (t < 16) {
        float bil = 0.f;
        for (int i = 0; i < 32; ++i) {
            float inner = 0.f;
            for (int j = 0; j < 32; ++j)
                inner += ntnW[(i * 32 + j) * 16 + t] * p2[j];
            bil += p1[i] * inner;
        }
        float blk = 0.f;
        for (int f = 0; f < 32; ++f)
            blk += ntnV[t * 64 + f] * p1[f] + ntnV[t * 64 + 32 + f] * p2[f];
        float sv = bil + blk + ntnb[t];
        feat[t] = sv > 0.f ? sv : 0.f;
        unsigned int hsum = 0;
        for (int j = 0; j < 16; ++j) hsum += ghist[b * 16 + j];
        feat[16 + t] = (float)ghist[b * 16 + t] / (float)(hsum ? hsum : 1u);
    }
    __syncthreads();
    if (t < 16) {
        float a = fc1b[t];
        for (int f = 0; f < 32; ++f) a += feat[f] * fc1W[f * 16 + t];
        h1[t] = a > 0.f ? a : 0.f;
    }
    __syncthreads();
    if (t == 0) {
        float a = scb[0];
        for (int j = 0; j < 16; ++j) a += h1[j] * scW[j];
        out[b] = 1.0f / (1.0f + __expf(-a));
    }
}

// ---------------------------------------------------------------------------
extern "C" void kernel_launch(void* const* d_in, const int* in_sizes, int n_in,
                              void* d_out, int out_size, void* d_ws, size_t ws_size,
                              hipStream_t stream)
{
    const float* x1  = (const float*)d_in[0];
    const float* x2  = (const float*)d_in[1];
    const int*   ei1 = (const int*)d_in[2];
    const int*   ei2 = (const int*)d_in[3];
    const float* W1  = (const float*)d_in[6];
    const float* b1  = (const float*)d_in[7];
    const float* W2  = (const float*)d_in[8];
    const float* b2  = (const float*)d_in[9];
    const float* W3  = (const float*)d_in[10];
    const float* b3  = (const float*)d_in[11];
    const float* attW = (const float*)d_in[12];
    const float* ntnW = (const float*)d_in[13];
    const float* ntnV = (const float*)d_in[14];
    const float* ntnb = (const float*)d_in[15];
    const float* fc1W = (const float*)d_in[16];
    const float* fc1b = (const float*)d_in[17];
    const float* scW  = (const float*)d_in[18];
    const float* scb  = (const float*)d_in[19];

    float* ws  = (float*)d_ws;
    float* deg = ws;                         // 65536
    float* dis = deg + TOT;                  // 65536 (rsqrt deg)
    float* inv = dis + TOT;                  // 65536 (1/deg)
    float* g   = inv + TOT;                  // 65536*64
    float* h   = g   + (size_t)TOT * 64;     // 65536*64
    float* af1 = h   + (size_t)TOT * 64;     // 65536*32
    float* af2 = af1 + (size_t)TOT * 32;     // 65536*32
    float* p1  = af2 + (size_t)TOT * 32;     // 128*32
    float* p2  = p1 + NB * 32;               // 128*32
    unsigned int* gmin  = (unsigned int*)(p2 + NB * 32); // 128
    unsigned int* gmax  = gmin + NB;                     // 128
    unsigned int* ghist = gmax + NB;                     // 128*16

    auto run_side = [&](const float* x, const int* ei, float* af, float* pooled) {
        const int* src = ei;
        const int* dst = ei + EE;
        fill_f32<<<TOT / 256, 256, 0, stream>>>(deg, 1.0f, TOT);   // self-loop
        deg_count<<<EE / 256, 256, 0, stream>>>(dst, deg, EE);
        deg_finalize<<<TOT / 256, 256, 0, stream>>>(deg, dis, inv, TOT);
        // layer 1: FIN=128 -> 64  (relu folded into layer-2 GEMM A-load)
        gemm_wmma<128, 64, false><<<512, 256, 0, stream>>>(x, W1, nullptr, g, TOT);
        gcn_self<64><<<TOT * 16 / 256, 256, 0, stream>>>(g, inv, b1, h, TOT);
        gcn_agg<64><<<EE * 16 / 256, 256, 0, stream>>>(src, dst, g, dis, h, EE);
        // layer 2: 64 -> 64
        gemm_wmma<64, 64, true><<<512, 256, 0, stream>>>(h, W2, nullptr, g, TOT);
        gcn_self<64><<<TOT * 16 / 256, 256, 0, stream>>>(g, inv, b2, h, TOT);
        gcn_agg<64><<<EE * 16 / 256, 256, 0, stream>>>(src, dst, g, dis, h, EE);
        // layer 3: 64 -> 32
        gemm_wmma<64, 32, true><<<512, 256, 0, stream>>>(h, W3, nullptr, g, TOT);
        gcn_self<32><<<TOT * 8 / 256, 256, 0, stream>>>(g, inv, b3, af, TOT);
        gcn_agg<32><<<EE * 8 / 256, 256, 0, stream>>>(src, dst, g, dis, af, EE);
        // attention pooling: hAtt = af @ attW (reuse g), then pool
        gemm_wmma<32, 32, false><<<512, 256, 0, stream>>>(af, attW, nullptr, g, TOT);
        att_pool<<<NB, 256, 0, stream>>>(af, g, pooled);
    };

    run_side(x1, ei1, af1, p1);
    run_side(x2, ei2, af2, p2);

    fill_u32<<<1, 256, 0, stream>>>(gmin, 0x7f800000u, NB);
    fill_u32<<<1, 256, 0, stream>>>(gmax, 0u, NB);
    fill_u32<<<8, 256, 0, stream>>>(ghist, 0u, NB * 16);
    sim_minmax<<<dim3(64, NB), 256, 0, stream>>>(af1, af2, gmin, gmax);
    sim_hist  <<<dim3(64, NB), 256, 0, stream>>>(af1, af2, gmin, gmax, ghist);

    ntn_final<<<NB, 64, 0, stream>>>(p1, p2, ghist, ntnW, ntnV, ntnb,
                                     fc1W, fc1b, scW, scb, (float*)d_out);
}